// GCN_10170482556975
// MI455X (gfx1250) — compile-verified
//
#include <hip/hip_runtime.h>
#include <cstdint>

// ---------------------------------------------------------------------------
// GCN (2-layer) for MI455X / gfx1250.
//   layer(A, W, b): u = dinv * (A @ W)    (WMMA f32 16x16x4)
//                   acc = u  (self-loop init)
//                   acc[d] += u[s]        (global_atomic_add_f32 per edge)
//                   out = dinv * acc + b  (+ relu for layer 1)
// dinv = rsqrt(deg) with deg = 1 + |incoming edges|, shared by both layers.
// ---------------------------------------------------------------------------

typedef __attribute__((ext_vector_type(2))) float v2f;
typedef __attribute__((ext_vector_type(8))) float v8f;

#define IN_C  128
#define HID_C 64
#define OUT_C 64

__device__ __forceinline__ void atomic_add_f32(float* p, float v) {
  __hip_atomic_fetch_add(p, v, __ATOMIC_RELAXED, __HIP_MEMORY_SCOPE_AGENT);
}

// deg[i] = 1.0 (self loop)
__global__ void k_deg_init(float* __restrict__ deg, int n) {
  int i = blockIdx.x * blockDim.x + threadIdx.x;
  if (i < n) deg[i] = 1.0f;
}

// deg[dst[e]] += 1
__global__ void k_deg_count(const long long* __restrict__ dst,
                            float* __restrict__ deg, int E) {
  int e = blockIdx.x * blockDim.x + threadIdx.x;
  if (e < E) atomic_add_f32(&deg[(int)dst[e]], 1.0f);
}

// in-place: deg -> rsqrt(deg)   (deg >= 1 always, matches reference's deg>0 path)
__global__ void k_rsqrt(float* __restrict__ deg, int n) {
  int i = blockIdx.x * blockDim.x + threadIdx.x;
  if (i < n) deg[i] = rsqrtf(deg[i]);
}

// One wave32 per 16x16 output tile. 4 waves per block cover the 64 columns.
// u[row,col] = dinv[row] * (A @ W)[row,col]; also duplicated into ACC as the
// self-loop initialization of the scatter accumulator.
template <int K>
__global__ __launch_bounds__(128) void k_gemm_wmma(
    const float* __restrict__ A,     // [N, K] row-major
    const float* __restrict__ W,     // [K, 64] row-major
    const float* __restrict__ dinv,  // [N]
    float* __restrict__ U,           // [N, 64]
    float* __restrict__ ACC)         // [N, 64]
{
  const int lane = threadIdx.x & 31;
  const int wave = threadIdx.x >> 5;
  const int r0 = blockIdx.x * 16;   // output row tile
  const int c0 = wave * 16;         // output col tile
  const int m  = lane & 15;         // M (for A) / N (for B) index
  const int kb = (lane >> 4) * 2;   // K sub-offset per lane half

  v8f acc = {0.f, 0.f, 0.f, 0.f, 0.f, 0.f, 0.f, 0.f};
  const float* arow = A + (size_t)(r0 + m) * K;

#pragma unroll
  for (int k = 0; k < K; k += 4) {
    v2f a, b;
    // A 16x4 fp32 layout: lane m holds rows; VGPR0/1 = K=kb, kb+1
    a.x = arow[k + kb];
    a.y = arow[k + kb + 1];
    // B 4x16 fp32 layout: lane m holds column c0+m; VGPR0/1 = K=kb, kb+1
    b.x = W[(size_t)(k + kb)     * 64 + c0 + m];
    b.y = W[(size_t)(k + kb + 1) * 64 + c0 + m];
    acc = __builtin_amdgcn_wmma_f32_16x16x4_f32(
        /*neg_a=*/false, a, /*neg_b=*/false, b,
        /*c_mod=*/(short)0, acc, /*reuse_a=*/false, /*reuse_b=*/false);
  }

  // D layout: VGPR j <- row (j + 8*(lane>>4)), col = lane&15
  const int col   = c0 + m;
  const int rbase = r0 + ((lane >> 4) << 3);
#pragma unroll
  for (int j = 0; j < 8; ++j) {
    const int row = rbase + j;
    const float v = dinv[row] * acc[j];
    U  [(size_t)row * 64 + col] = v;
    ACC[(size_t)row * 64 + col] = v;
  }
}

// 16 threads per edge; each thread moves 4 channels (float4 gather from L2-
// resident U, 4 native f32 atomic adds into ACC).
__global__ __launch_bounds__(256) void k_scatter(
    const long long* __restrict__ src, const long long* __restrict__ dst,
    const float* __restrict__ U, float* __restrict__ ACC, int E)
{
  const long long tid = (long long)blockIdx.x * blockDim.x + threadIdx.x;
  const int e = (int)(tid >> 4);
  if (e >= E) return;
  const int q = (int)(tid & 15);
  const int s = (int)src[e];
  const int d = (int)dst[e];
  const float4 v = *reinterpret_cast<const float4*>(U + (size_t)s * 64 + q * 4);
  float* p = ACC + (size_t)d * 64 + q * 4;
  atomic_add_f32(p + 0, v.x);
  atomic_add_f32(p + 1, v.y);
  atomic_add_f32(p + 2, v.z);
  atomic_add_f32(p + 3, v.w);
}

// out[i,c] = (relu?) dinv[i] * acc[i,c] + bias[c]    (acc==out in-place is OK)
__global__ void k_finalize(const float* __restrict__ acc,
                           const float* __restrict__ dinv,
                           const float* __restrict__ bias,
                           float* __restrict__ out, int n, int relu)
{
  const int i = blockIdx.x * blockDim.x + threadIdx.x;
  if (i >= n * 64) return;
  const int row = i >> 6;
  const int c = i & 63;
  float v = dinv[row] * acc[i] + bias[c];
  if (relu) v = fmaxf(v, 0.0f);
  out[i] = v;
}

extern "C" void kernel_launch(void* const* d_in, const int* in_sizes, int n_in,
                              void* d_out, int out_size, void* d_ws, size_t ws_size,
                              hipStream_t stream) {
  const float*     x    = (const float*)d_in[0];
  const long long* edge = (const long long*)d_in[1];  // int64 [2, E]
  const float*     W1   = (const float*)d_in[2];
  const float*     b1   = (const float*)d_in[3];
  const float*     W2   = (const float*)d_in[4];
  const float*     b2   = (const float*)d_in[5];

  const int N = in_sizes[0] / IN_C;  // 50000 (divisible by 16)
  const int E = in_sizes[1] / 2;     // 1600000
  const long long* src = edge;
  const long long* dst = edge + E;

  // Workspace layout: dinv[N] | U[N*64] | ACC1/H[N*64]   (~26 MB)
  char* ws = (char*)d_ws;
  size_t off = 0;
  float* dinv = (float*)(ws + off); off += (((size_t)N * 4) + 255) & ~(size_t)255;
  float* U    = (float*)(ws + off); off += (((size_t)N * 64 * 4) + 255) & ~(size_t)255;
  float* ACC1 = (float*)(ws + off);
  float* out  = (float*)d_out;

  // ---- shared normalization ----
  k_deg_init <<<(N + 255) / 256, 256, 0, stream>>>(dinv, N);
  k_deg_count<<<(E + 255) / 256, 256, 0, stream>>>(dst, dinv, E);
  k_rsqrt    <<<(N + 255) / 256, 256, 0, stream>>>(dinv, N);

  const long long sthreads = (long long)E * 16;
  const unsigned  sblocks  = (unsigned)((sthreads + 255) / 256);

  // ---- layer 1: x[ N x128 ] @ W1 -> 64, relu ----
  k_gemm_wmma<IN_C><<<N / 16, 128, 0, stream>>>(x, W1, dinv, U, ACC1);
  k_scatter<<<sblocks, 256, 0, stream>>>(src, dst, U, ACC1, E);
  k_finalize<<<(N * 64 + 255) / 256, 256, 0, stream>>>(ACC1, dinv, b1, ACC1, N, 1);

  // ---- layer 2: h[ N x64 ] @ W2 -> 64 ; accumulator lives in d_out ----
  k_gemm_wmma<HID_C><<<N / 16, 128, 0, stream>>>(ACC1, W2, dinv, U, out);
  k_scatter<<<sblocks, 256, 0, stream>>>(src, dst, U, out, E);
  k_finalize<<<(N * 64 + 255) / 256, 256, 0, stream>>>(out, dinv, b2, out, N, 0);
}